// InEKFLevel15_NoMeas_51170240365111
// MI455X (gfx1250) — compile-verified
//
#include <hip/hip_runtime.h>
#include <math.h>

#define D_MODEL 1024
#define HIDDEN  128
#define NCOL    512              // H * NB
#define B_      4
#define L_      4096
#define M_ROWS  16384            // B * L
#define NTOT    ((size_t)M_ROWS * NCOL)

typedef __attribute__((ext_vector_type(2))) float v2f;
typedef __attribute__((ext_vector_type(4))) float v4f;
typedef __attribute__((ext_vector_type(8))) float v8f;
typedef __attribute__((ext_vector_type(4))) int   v4i;

// GEMM1 tiling
#define BM     64                // rows per block (4 x 16 M-tiles per wave)
#define BK     32                // K-chunk staged in LDS
#define APAD   36                // row stride in floats: 16B-aligned, conflict-free
#define NCHUNK (D_MODEL / BK)    // 32

#if __has_builtin(__builtin_amdgcn_global_load_async_to_lds_b128) && \
    __has_builtin(__builtin_amdgcn_s_wait_asynccnt)
#define HAVE_ASYNC_LDS 1
typedef __attribute__((address_space(1))) v4i g_v4i;   // global int4*
typedef __attribute__((address_space(3))) v4i l_v4i;   // LDS int4*
#else
#define HAVE_ASYNC_LDS 0
#endif

__device__ __forceinline__ float gelu_exact(float x) {
    return 0.5f * x * (1.0f + erff(x * 0.7071067811865476f));
}

// ---------------------------------------------------------------------------
// Kernel 1: h = GELU(content_emb @ W1 + b1)   [16384,1024] x [1024,128]
// Block: 256 threads / 8 waves. Block tile: 64 rows x 128 cols.
// A tile double-buffered in LDS via async global->LDS loads; each wave does
// 4 M-tiles sharing one B fragment per K-step (V_WMMA_F32_16X16X4_F32).
// ---------------------------------------------------------------------------
__global__ __launch_bounds__(256) void gemm1_gelu_kernel(
    const float* __restrict__ A,    // [M, 1024]
    const float* __restrict__ W1,   // [1024, 128]
    const float* __restrict__ b1,   // [128]
    float* __restrict__ Hout)       // [M, 128]
{
    __shared__ float Abuf[2][BM * APAD];   // 2 x 9216 B

    const int tid  = threadIdx.x;
    const int lane = tid & 31;
    const int wave = tid >> 5;               // 0..7 -> N tile
    const int row0 = blockIdx.x * BM;
    const int col0 = wave * 16;
    const int m    = lane & 15;
    const int half = lane >> 4;              // lane group 0/1
    const int n    = m;

    const float* Abase = A + (size_t)row0 * D_MODEL;

    // -- cooperative fill of one K-chunk into LDS buffer s ------------------
    auto fill_issue = [&](int s, int c) {
        const float* Ablk = Abase + c * BK;
#pragma unroll
        for (int r = 0; r < 2; ++r) {
            const int slot = tid + 256 * r;        // 512 float4 slots
            const int row  = slot >> 3;            // 64 rows
            const int q    = slot & 7;             // 8 float4 per row
            const float* g = Ablk + (size_t)row * D_MODEL + q * 4;
            float*       l = &Abuf[s][row * APAD + q * 4];
#if HAVE_ASYNC_LDS
            __builtin_amdgcn_global_load_async_to_lds_b128(
                (g_v4i*)g, (l_v4i*)l, 0, 0);
#else
            *(v4f*)l = *(const v4f*)g;
#endif
        }
    };

    v8f acc[4] = {};

    fill_issue(0, 0);
    for (int c = 0; c < NCHUNK; ++c) {
        const int s = c & 1;
        if (c + 1 < NCHUNK) {
            fill_issue(s ^ 1, c + 1);              // next chunk in flight
#if HAVE_ASYNC_LDS
            __builtin_amdgcn_s_wait_asynccnt(2);   // current chunk landed
#endif
        } else {
#if HAVE_ASYNC_LDS
            __builtin_amdgcn_s_wait_asynccnt(0);
#endif
        }
        __syncthreads();

        // hoist the 8 B fragments for this chunk (clustered global loads)
        v2f bfr[8];
#pragma unroll
        for (int j = 0; j < 8; ++j) {
            const int kg = c * BK + j * 4 + 2 * half;
            const float* bp = W1 + (size_t)kg * HIDDEN + col0 + n;
            bfr[j].x = bp[0];
            bfr[j].y = bp[HIDDEN];
        }

#pragma unroll
        for (int j = 0; j < 8; ++j) {
#pragma unroll
            for (int mt = 0; mt < 4; ++mt) {
                const int row = mt * 16 + m;
                const v2f a = *(const v2f*)&Abuf[s][row * APAD + j * 4 + 2 * half];
                acc[mt] = __builtin_amdgcn_wmma_f32_16x16x4_f32(
                              false, a, false, bfr[j], (short)0, acc[mt],
                              false, false);
            }
        }
        __syncthreads();
    }

    const float bias = b1[col0 + n];
#pragma unroll
    for (int mt = 0; mt < 4; ++mt) {
#pragma unroll
        for (int v = 0; v < 8; ++v) {
            const int mr = mt * 16 + v + 8 * half;   // C layout: m = v + 8*(lane>=16)
            Hout[(size_t)(row0 + mr) * HIDDEN + col0 + n] =
                gelu_exact(acc[mt][v] + bias);
        }
    }
}

// ---------------------------------------------------------------------------
// Kernel 2: y = h @ W2 + b2; logR = clip(y,-5,5); R = exp(logR);
//           Pi = exp(log_Pi); K = Pi / max(Pi+R, 1e-8)
// Block tile: 64 rows x 128 cols (grid.y covers 512 cols); 4 M-tiles/wave
// share each B fragment. Writes Pi/K/R outputs directly.
// ---------------------------------------------------------------------------
__global__ __launch_bounds__(256) void gemm2_epilogue_kernel(
    const float* __restrict__ Hin,     // [M, 128]
    const float* __restrict__ W2,      // [128, 512]
    const float* __restrict__ b2,      // [512]
    const float* __restrict__ log_Pi,  // [512]
    float* __restrict__ outPi,
    float* __restrict__ outK,
    float* __restrict__ outR)
{
    const int lane = threadIdx.x & 31;
    const int wave = threadIdx.x >> 5;
    const int row0 = blockIdx.x * BM;
    const int col0 = blockIdx.y * 128 + wave * 16;
    const int m    = lane & 15;
    const int half = lane >> 4;
    const int n    = m;

    v8f acc[4] = {};
    for (int k = 0; k < HIDDEN; k += 4) {
        const float* bp = W2 + (size_t)(k + 2 * half) * NCOL + col0 + n;
        v2f b;
        b.x = bp[0];
        b.y = bp[NCOL];
#pragma unroll
        for (int mt = 0; mt < 4; ++mt) {
            const float* ap =
                Hin + (size_t)(row0 + mt * 16 + m) * HIDDEN + k + 2 * half;
            v2f a;
            a.x = ap[0];
            a.y = ap[1];
            acc[mt] = __builtin_amdgcn_wmma_f32_16x16x4_f32(
                          false, a, false, b, (short)0, acc[mt], false, false);
        }
    }

    const int   col  = col0 + n;
    const float bias = b2[col];
    const float Pi   = expf(log_Pi[col]);

#pragma unroll
    for (int mt = 0; mt < 4; ++mt) {
#pragma unroll
        for (int v = 0; v < 8; ++v) {
            const int    mr  = mt * 16 + v + 8 * half;
            const size_t idx = (size_t)(row0 + mr) * NCOL + col;
            float y = acc[mt][v] + bias;
            y = fminf(5.0f, fmaxf(-5.0f, y));
            const float R  = expf(y);
            const float Kg = Pi / fmaxf(Pi + R, 1e-8f);
            outPi[idx] = Pi;
            outK[idx]  = Kg;
            outR[idx]  = R;
        }
    }
}

// ---------------------------------------------------------------------------
// Kernel 3: affine scan d_t = alpha_t * d_{t-1} + u_t over L per (b,channel).
// alpha = 1-K, u = K * wrap(-theta). 2048 blocks (one per channel), 256
// threads x 16-element segments, block Hillis-Steele scan of (a,u) in LDS.
// ---------------------------------------------------------------------------
#define SEG 16
__global__ __launch_bounds__(256) void affine_scan_kernel(
    const float* __restrict__ theta,
    const float* __restrict__ Kmat,
    float* __restrict__ out0)
{
    const int b  = blockIdx.x / NCOL;
    const int ch = blockIdx.x % NCOL;
    const int t  = threadIdx.x;
    const size_t base = (size_t)b * L_ * NCOL + ch;

    float a_loc[SEG], u_loc[SEG], th_loc[SEG];
    float a_acc = 1.0f, u_acc = 0.0f;
#pragma unroll
    for (int i = 0; i < SEG; ++i) {
        const size_t idx = base + (size_t)(t * SEG + i) * NCOL;
        const float th = theta[idx];
        const float Kg = Kmat[idx];
        const float nu = atan2f(sinf(-th), cosf(-th));
        const float al = 1.0f - Kg;
        const float uu = Kg * nu;
        a_loc[i] = al; u_loc[i] = uu; th_loc[i] = th;
        u_acc = al * u_acc + uu;
        a_acc = al * a_acc;
    }

    __shared__ float sa[256];
    __shared__ float su[256];
    sa[t] = a_acc;
    su[t] = u_acc;
    __syncthreads();

    for (int off = 1; off < 256; off <<= 1) {
        float pa = 1.0f, pu = 0.0f;
        const bool valid = (t >= off);
        if (valid) { pa = sa[t - off]; pu = su[t - off]; }
        __syncthreads();
        if (valid) {
            const float ac = sa[t], uc = su[t];
            sa[t] = ac * pa;
            su[t] = ac * pu + uc;
        }
        __syncthreads();
    }

    float d = (t > 0) ? su[t - 1] : 0.0f;
#pragma unroll
    for (int i = 0; i < SEG; ++i) {
        d = a_loc[i] * d + u_loc[i];
        const size_t idx = base + (size_t)(t * SEG + i) * NCOL;
        out0[idx] = th_loc[i] + d;
    }
}

// ---------------------------------------------------------------------------
extern "C" void kernel_launch(void* const* d_in, const int* in_sizes, int n_in,
                              void* d_out, int out_size, void* d_ws, size_t ws_size,
                              hipStream_t stream) {
    const float* theta  = (const float*)d_in[0];
    const float* emb    = (const float*)d_in[1];
    const float* log_Pi = (const float*)d_in[2];
    const float* W1     = (const float*)d_in[3];
    const float* b1     = (const float*)d_in[4];
    const float* W2     = (const float*)d_in[5];
    const float* b2     = (const float*)d_in[6];

    float* out   = (float*)d_out;
    float* out0  = out;
    float* outPi = out + NTOT;
    float* outK  = out + 2 * NTOT;
    float* outR  = out + 3 * NTOT;

    float* Hws = (float*)d_ws;   // [16384,128] fp32 = 8 MiB scratch

    gemm1_gelu_kernel<<<dim3(M_ROWS / BM), 256, 0, stream>>>(emb, W1, b1, Hws);
    gemm2_epilogue_kernel<<<dim3(M_ROWS / BM, NCOL / 128), 256, 0, stream>>>(
        Hws, W2, b2, log_Pi, outPi, outK, outR);
    affine_scan_kernel<<<dim3(B_ * NCOL), 256, 0, stream>>>(theta, outK, out0);
}